// SparseLinear2_4415226380844
// MI455X (gfx1250) — compile-verified
//
#include <hip/hip_runtime.h>
#include <hip/hip_bf16.h>
#include <stdint.h>

#define NUM_INPUTS_  50000
#define NUM_OUTPUTS_ 5000
#define NNZ_TOTAL_   1000000
#define BATCH_       128

#define WAVES_PER_BLOCK 8
#define CHUNK 32              // nnz staged per wave per stage
#define STAGES 4
#define NNZ_PER_WAVE (CHUNK * STAGES)   // 128

// ---------------- CDNA5 async global->LDS helpers (inline asm) ----------------

__device__ __forceinline__ uint32_t lds_off(const void* p) {
  // Generic pointers into LDS carry the LDS byte offset in the low 32 bits.
  return (uint32_t)(uintptr_t)p;
}

__device__ __forceinline__ void async_ld_b64(uint32_t lds, uint32_t voff, uint64_t saddr) {
  asm volatile("global_load_async_to_lds_b64 %0, %1, %2"
               :: "v"(lds), "v"(voff), "s"(saddr) : "memory");
}

__device__ __forceinline__ void async_ld_b32(uint32_t lds, uint32_t voff, uint64_t saddr) {
  asm volatile("global_load_async_to_lds_b32 %0, %1, %2"
               :: "v"(lds), "v"(voff), "s"(saddr) : "memory");
}

#define S_WAIT_ASYNCCNT(n) asm volatile("s_wait_asynccnt " #n ::: "memory")

__device__ __forceinline__ void atomic_add_f32(float* p, float v) {
  __hip_atomic_fetch_add(p, v, __ATOMIC_RELAXED, __HIP_MEMORY_SCOPE_AGENT);
}

// ---------------- Kernel: main scatter with async-LDS streamed metadata ------
// One wave per 128 nnz. 32 lanes * float4 == one 512B batch-row of xT / yT.
// (Kept first in the file so the disasm snippet shows this kernel.)

__global__ void __launch_bounds__(256)
spmm_scatter_kernel(const float* __restrict__ xT,      // [NUM_INPUTS][128]
                    const int*   __restrict__ conn,    // [NNZ][2] (row, col)
                    const float* __restrict__ w,       // [NNZ]
                    float*       __restrict__ yT) {    // [NUM_OUTPUTS][128]
  __shared__ int2  s_conn[WAVES_PER_BLOCK][2][CHUNK];
  __shared__ float s_w   [WAVES_PER_BLOCK][2][CHUNK];

  const int wave = threadIdx.x >> 5;
  const int lane = threadIdx.x & 31;
  const long long waveGlobal = (long long)blockIdx.x * WAVES_PER_BLOCK + wave;
  const long long kBase = waveGlobal * NNZ_PER_WAVE;
  if (kBase >= NNZ_TOTAL_) return;   // uniform per wave

  const uint64_t connBase = (uint64_t)(uintptr_t)conn;
  const uint64_t wBase    = (uint64_t)(uintptr_t)w;

  const uint32_t lds_c0 = lds_off(&s_conn[wave][0][lane]);
  const uint32_t lds_c1 = lds_off(&s_conn[wave][1][lane]);
  const uint32_t lds_w0 = lds_off(&s_w[wave][0][lane]);
  const uint32_t lds_w1 = lds_off(&s_w[wave][1][lane]);

  // issue stage s into buffer buf (clamped addresses for the tail wave)
  auto issue = [&](int s, int buf) {
    long long k = kBase + (long long)s * CHUNK + lane;
    if (k > NNZ_TOTAL_ - 1) k = NNZ_TOTAL_ - 1;
    uint32_t kk = (uint32_t)k;
    async_ld_b64(buf ? lds_c1 : lds_c0, kk * 8u, connBase);
    async_ld_b32(buf ? lds_w1 : lds_w0, kk * 4u, wBase);
  };

  issue(0, 0);

  const float4* __restrict__ xT4 = (const float4*)xT;

  for (int s = 0; s < STAGES; ++s) {
    const int buf = s & 1;
    if (s + 1 < STAGES) {
      issue(s + 1, buf ^ 1);
      S_WAIT_ASYNCCNT(2);      // the 2 ops of the current stage are complete
    } else {
      S_WAIT_ASYNCCNT(0);
    }

    for (int j = 0; j < CHUNK; ++j) {
      long long k = kBase + (long long)s * CHUNK + j;
      if (k >= NNZ_TOTAL_) break;                 // uniform across wave
      volatile int*   cp = (volatile int*)&s_conn[wave][buf][j];
      volatile float* wp = (volatile float*)&s_w[wave][buf][j];
      const int   row = cp[0];
      const int   col = cp[1];
      const float wt  = *wp;

      float4 xv = xT4[(size_t)row * (BATCH_ / 4) + lane];   // coalesced 512B
      float* dst = yT + (size_t)col * BATCH_ + lane * 4;    // coalesced 512B
      atomic_add_f32(dst + 0, xv.x * wt);
      atomic_add_f32(dst + 1, xv.y * wt);
      atomic_add_f32(dst + 2, xv.z * wt);
      atomic_add_f32(dst + 3, xv.w * wt);
    }
  }
}

// ---------------- Kernel: transpose x (128 x 50000) -> xT (50000 x 128) ------

__global__ void __launch_bounds__(256)
transpose_x_kernel(const float* __restrict__ x, float* __restrict__ xT) {
  __shared__ float tile[32][33];
  const int tx = threadIdx.x;      // 0..31
  const int ty = threadIdx.y;      // 0..7
  const int rowBlk = blockIdx.x * 32;   // over NUM_INPUTS
  const int bBlk   = blockIdx.y * 32;   // over BATCH (4 blocks)

  #pragma unroll
  for (int i = 0; i < 4; ++i) {
    int b = bBlk + ty + i * 8;
    int r = rowBlk + tx;
    float v = (r < NUM_INPUTS_) ? x[(size_t)b * NUM_INPUTS_ + r] : 0.0f;
    tile[ty + i * 8][tx] = v;      // tile[b_local][r_local]
  }
  __syncthreads();
  #pragma unroll
  for (int i = 0; i < 4; ++i) {
    int r = rowBlk + ty + i * 8;
    int b = bBlk + tx;
    if (r < NUM_INPUTS_)
      xT[(size_t)r * BATCH_ + b] = tile[tx][ty + i * 8];
  }
}

// ---------------- Kernel: out[b,c] = yT[c,b] + bias[c] -----------------------

__global__ void __launch_bounds__(256)
finalize_kernel(const float* __restrict__ yT, const float* __restrict__ bias,
                float* __restrict__ out) {
  int tid = blockIdx.x * blockDim.x + threadIdx.x;
  if (tid < BATCH_ * NUM_OUTPUTS_) {
    int b = tid / NUM_OUTPUTS_;
    int c = tid - b * NUM_OUTPUTS_;
    out[tid] = yT[(size_t)c * BATCH_ + b] + bias[c];
  }
}

// ---------------- Fallback path (if workspace is too small) ------------------

__global__ void __launch_bounds__(256)
init_out_kernel(float* __restrict__ out, const float* __restrict__ bias) {
  int tid = blockIdx.x * blockDim.x + threadIdx.x;
  if (tid < BATCH_ * NUM_OUTPUTS_) {
    out[tid] = bias[tid % NUM_OUTPUTS_];
  }
}

__global__ void __launch_bounds__(256)
spmm_scatter_slow_kernel(const float* __restrict__ x, const int* __restrict__ conn,
                         const float* __restrict__ w, float* __restrict__ out) {
  const int wave = threadIdx.x >> 5;
  const int lane = threadIdx.x & 31;
  const long long k = (long long)blockIdx.x * WAVES_PER_BLOCK + wave;
  if (k >= NNZ_TOTAL_) return;
  const int row = conn[2 * k];
  const int col = conn[2 * k + 1];
  const float wt = w[k];
  #pragma unroll
  for (int i = 0; i < 4; ++i) {
    int b = lane + 32 * i;
    float v = x[(size_t)b * NUM_INPUTS_ + row] * wt;
    atomic_add_f32(&out[(size_t)b * NUM_OUTPUTS_ + col], v);
  }
}

// ---------------- Host launch -------------------------------------------------

extern "C" void kernel_launch(void* const* d_in, const int* in_sizes, int n_in,
                              void* d_out, int out_size, void* d_ws, size_t ws_size,
                              hipStream_t stream) {
  const float* x    = (const float*)d_in[0];   // [128, 50000] f32
  const int*   conn = (const int*)d_in[1];     // [1e6, 2] int
  const float* w    = (const float*)d_in[2];   // [1e6] f32
  const float* bias = (const float*)d_in[3];   // [5000] f32
  float* out = (float*)d_out;                  // [128, 5000] f32

  const size_t xT_bytes = (size_t)NUM_INPUTS_ * BATCH_ * sizeof(float);   // 25.6 MB
  const size_t yT_bytes = (size_t)NUM_OUTPUTS_ * BATCH_ * sizeof(float);  // 2.56 MB

  if (ws_size >= xT_bytes + yT_bytes) {
    float* xT = (float*)d_ws;
    float* yT = (float*)((char*)d_ws + xT_bytes);

    dim3 tb(32, 8);
    dim3 tg((NUM_INPUTS_ + 31) / 32, BATCH_ / 32);
    transpose_x_kernel<<<tg, tb, 0, stream>>>(x, xT);

    hipMemsetAsync(yT, 0, yT_bytes, stream);

    const int totalWaves = (NNZ_TOTAL_ + NNZ_PER_WAVE - 1) / NNZ_PER_WAVE;  // 7813
    const int blocks = (totalWaves + WAVES_PER_BLOCK - 1) / WAVES_PER_BLOCK;
    spmm_scatter_kernel<<<blocks, 256, 0, stream>>>(xT, conn, w, yT);

    const int n = BATCH_ * NUM_OUTPUTS_;
    finalize_kernel<<<(n + 255) / 256, 256, 0, stream>>>(yT, bias, out);
  } else {
    const int n = BATCH_ * NUM_OUTPUTS_;
    init_out_kernel<<<(n + 255) / 256, 256, 0, stream>>>(out, bias);
    const int blocks = (NNZ_TOTAL_ + WAVES_PER_BLOCK - 1) / WAVES_PER_BLOCK;
    spmm_scatter_slow_kernel<<<blocks, 256, 0, stream>>>(x, conn, w, out);
  }
}